// GCN_base_19825569038949
// MI455X (gfx1250) — compile-verified
//
#include <hip/hip_runtime.h>
#include <hip/hip_bf16.h>

typedef __attribute__((ext_vector_type(2))) float v2f;
typedef __attribute__((ext_vector_type(8))) float v8f;

// ---------------------------------------------------------------------------
// Degree / normalization kernels
// ---------------------------------------------------------------------------
__global__ void deg_init_kernel(float* __restrict__ deg, int n) {
    int i = blockIdx.x * blockDim.x + threadIdx.x;
    if (i < n) deg[i] = 1.0f;  // self-loop contributes 1 to every node's degree
}

__global__ void deg_accum_kernel(const long long* __restrict__ col,
                                 float* __restrict__ deg, int E) {
    int e = blockIdx.x * blockDim.x + threadIdx.x;
    if (e < E) atomicAdd(&deg[(int)col[e]], 1.0f);
}

__global__ void dinv_kernel(float* __restrict__ deg, int n) {
    int i = blockIdx.x * blockDim.x + threadIdx.x;
    if (i < n) {
        float d = deg[i];
        deg[i] = (d > 0.0f) ? rsqrtf(d) : 0.0f;  // in-place -> dinv
    }
}

// ---------------------------------------------------------------------------
// fp32 WMMA GEMM:  C[M,N] = (RELU ? relu(A) : A)[M,K] @ B[K,N]
// One wave computes TWO 16x16 C tiles stacked in M (32 rows x 16 cols),
// reusing one B fragment for two V_WMMA_F32_16X16X4_F32 per K-step.
// Block = 128 threads (4 waves) -> 32(M) x 64(N) tile per block.
// Fragment layout (ISA 7.12.2, 32-bit A 16x4 / B 4x16):
//   lanes 0-15  : K = k+0 (v0), k+1 (v1)
//   lanes 16-31 : K = k+2 (v0), k+3 (v1)
//   C/D: lane r (0-15) = column n0+r; VGPR i = row m0+i (lanes 0-15) /
//        m0+8+i (lanes 16-31)
// ---------------------------------------------------------------------------
template <bool RELU>
__global__ void __launch_bounds__(128)
gcn_gemm_wmma_f32(const float* __restrict__ A, const float* __restrict__ B,
                  float* __restrict__ C, int M, int K, int N) {
    const int lane = threadIdx.x & 31;
    const int wave = threadIdx.x >> 5;
    const int r    = lane & 15;
    const int half = lane >> 4;

    const int n0 = blockIdx.x * 64 + wave * 16;
    const int m0 = blockIdx.y * 32;           // two stacked M-tiles

    int ar0 = m0 + r;
    int ar1 = m0 + 16 + r;
    if (ar0 >= M) ar0 = M - 1;                // clamp (tail tile)
    if (ar1 >= M) ar1 = M - 1;
    const float* __restrict__ arow0 = A + (size_t)ar0 * K + 2 * half;
    const float* __restrict__ arow1 = A + (size_t)ar1 * K + 2 * half;
    const float* __restrict__ bcol  = B + (size_t)(2 * half) * N + (n0 + r);

    v8f acc0 = {0.f, 0.f, 0.f, 0.f, 0.f, 0.f, 0.f, 0.f};
    v8f acc1 = {0.f, 0.f, 0.f, 0.f, 0.f, 0.f, 0.f, 0.f};

#pragma unroll 4
    for (int k = 0; k < K; k += 4) {
        v2f b;
        b.x = bcol[(size_t)(k + 0) * N];
        b.y = bcol[(size_t)(k + 1) * N];

        v2f a0, a1;
        a0.x = arow0[k + 0];
        a0.y = arow0[k + 1];
        a1.x = arow1[k + 0];
        a1.y = arow1[k + 1];
        if (RELU) {
            a0.x = fmaxf(a0.x, 0.0f);
            a0.y = fmaxf(a0.y, 0.0f);
            a1.x = fmaxf(a1.x, 0.0f);
            a1.y = fmaxf(a1.y, 0.0f);
        }
        // D = A*B + C   (8 args: neg_a, A, neg_b, B, c_mod, C, reuse_a, reuse_b)
        acc0 = __builtin_amdgcn_wmma_f32_16x16x4_f32(
            false, a0, false, b, (short)0, acc0, false, false);
        acc1 = __builtin_amdgcn_wmma_f32_16x16x4_f32(
            false, a1, false, b, (short)0, acc1, false, false);
    }

    float* __restrict__ crow0 = C + (size_t)(m0 + 8 * half) * N + n0 + r;
    float* __restrict__ crow1 = C + (size_t)(m0 + 16 + 8 * half) * N + n0 + r;
#pragma unroll
    for (int i = 0; i < 8; ++i) {
        int mrow0 = m0 + 8 * half + i;
        int mrow1 = m0 + 16 + 8 * half + i;
        if (mrow0 < M) crow0[(size_t)i * N] = acc0[i];
        if (mrow1 < M) crow1[(size_t)i * N] = acc1[i];
    }
}

// ---------------------------------------------------------------------------
// Aggregation: self-loop init  g[i,:] = t[i,:] * dinv[i]^2   (no atomics)
// ---------------------------------------------------------------------------
__global__ void self_init_kernel(const float* __restrict__ t,
                                 const float* __restrict__ dinv,
                                 float* __restrict__ g, int total /* = N*128 */) {
    int tid = blockIdx.x * blockDim.x + threadIdx.x;
    if (tid < total) {
        int i = tid >> 7;  // H = 128
        float di = dinv[i];
        g[tid] = t[tid] * di * di;
    }
}

// ---------------------------------------------------------------------------
// Edge scatter: one wave per edge, each lane handles 4 features.
// g[col,:] += t[row,:] * dinv[row]*dinv[col]
// ---------------------------------------------------------------------------
__global__ void scatter_kernel(const float* __restrict__ t,
                               const long long* __restrict__ row,
                               const long long* __restrict__ col,
                               const float* __restrict__ dinv,
                               float* __restrict__ g, int E) {
    long long tid = (long long)blockIdx.x * blockDim.x + threadIdx.x;
    int e = (int)(tid >> 5);
    if (e >= E) return;
    int q = (int)(tid & 31);           // feature chunk: 4 floats per lane

    int rsrc = (int)row[e];
    int cdst = (int)col[e];
    float nm = dinv[rsrc] * dinv[cdst];

    const float4 v = *(const float4*)(t + (size_t)rsrc * 128 + q * 4);
    float* dst = g + (size_t)cdst * 128 + q * 4;
    atomicAdd(dst + 0, v.x * nm);
    atomicAdd(dst + 1, v.y * nm);
    atomicAdd(dst + 2, v.z * nm);
    atomicAdd(dst + 3, v.w * nm);
}

// ---------------------------------------------------------------------------
// Launch: 3-layer GCN pipeline
// ---------------------------------------------------------------------------
extern "C" void kernel_launch(void* const* d_in, const int* in_sizes, int n_in,
                              void* d_out, int out_size, void* d_ws, size_t ws_size,
                              hipStream_t stream) {
    const float*      x  = (const float*)d_in[0];
    const long long*  ei = (const long long*)d_in[1];   // int64 edge_index [2,E]
    const float*      W0 = (const float*)d_in[2];       // [256,128]
    const float*      W1 = (const float*)d_in[3];       // [128,128]
    const float*      W2 = (const float*)d_in[4];       // [128,128]

    const int H  = 128;
    const int K0 = 256;
    const int Nn = in_sizes[0] / K0;        // 50000 nodes
    const int E  = in_sizes[1] / 2;         // 800000 edges
    const long long* row = ei;
    const long long* col = ei + E;

    // Workspace layout: [dinv | t (gemm out) | g (agg out)]
    float* dinv = (float*)d_ws;
    size_t off  = ((size_t)Nn * sizeof(float) + 255) & ~(size_t)255;
    float* tbuf = (float*)((char*)d_ws + off);
    size_t hbytes = ((size_t)Nn * H * sizeof(float) + 255) & ~(size_t)255;
    float* gbuf = (float*)((char*)d_ws + off + hbytes);
    float* outp = (float*)d_out;

    const int B256 = 256;
    int mtiles2 = (Nn + 31) / 32;           // two 16-row tiles per wave
    dim3 ggrid(H / 64, mtiles2);            // 4 waves cover 64 cols
    dim3 gblk(128);
    int selfGrid = (Nn * H + B256 - 1) / B256;
    long long sthreads = (long long)E * 32;
    int scatGrid = (int)((sthreads + B256 - 1) / B256);

    // --- normalization ---
    deg_init_kernel<<<(Nn + B256 - 1) / B256, B256, 0, stream>>>(dinv, Nn);
    deg_accum_kernel<<<(E + B256 - 1) / B256, B256, 0, stream>>>(col, dinv, E);
    dinv_kernel<<<(Nn + B256 - 1) / B256, B256, 0, stream>>>(dinv, Nn);

    // --- layer 1:  g = Agg(x @ W0)        (ReLU deferred into next GEMM) ---
    gcn_gemm_wmma_f32<false><<<ggrid, gblk, 0, stream>>>(x, W0, tbuf, Nn, K0, H);
    self_init_kernel<<<selfGrid, B256, 0, stream>>>(tbuf, dinv, gbuf, Nn * H);
    scatter_kernel<<<scatGrid, B256, 0, stream>>>(tbuf, row, col, dinv, gbuf, E);

    // --- layer 2:  g = Agg(relu(g) @ W1) ---
    gcn_gemm_wmma_f32<true><<<ggrid, gblk, 0, stream>>>(gbuf, W1, tbuf, Nn, H, H);
    self_init_kernel<<<selfGrid, B256, 0, stream>>>(tbuf, dinv, gbuf, Nn * H);
    scatter_kernel<<<scatGrid, B256, 0, stream>>>(tbuf, row, col, dinv, gbuf, E);

    // --- layer 3:  out = Agg(relu(g) @ W2) ---
    gcn_gemm_wmma_f32<true><<<ggrid, gblk, 0, stream>>>(gbuf, W2, tbuf, Nn, H, H);
    self_init_kernel<<<selfGrid, B256, 0, stream>>>(tbuf, dinv, outp, Nn * H);
    scatter_kernel<<<scatGrid, B256, 0, stream>>>(tbuf, row, col, dinv, outp, E);
}